// ForwardLoss_25761213841995
// MI455X (gfx1250) — compile-verified
//
#include <hip/hip_runtime.h>

// Problem constants (from the reference)
#define B_   256
#define K_   500
#define W_   272
#define HW_  (152 * 272)
#define TBL  1024            // hash table slots (load factor ~0.49)
#define NT   256             // threads per block (8 wave32s)

// ---------------------------------------------------------------------------
// gfx1250 helpers: async global->LDS copy (ASYNCcnt path) via inline asm so it
// works on both ROCm-7.2 (clang-22) and upstream clang-23 toolchains.
// ---------------------------------------------------------------------------
typedef __attribute__((address_space(3))) void lds_void_t;

__device__ __forceinline__ unsigned lds_addr32(void* p) {
  // generic -> LDS addrspace cast, then take the 32-bit LDS byte offset
  return (unsigned)(unsigned long long)(lds_void_t*)p;
}

__device__ __forceinline__ void async_copy_dword(void* lds_dst, const void* gsrc) {
#if defined(__gfx1250__)
  unsigned dst = lds_addr32(lds_dst);
  unsigned long long src = (unsigned long long)gsrc;
  // GLOBAL_LOAD_ASYNC_TO_LDS_B32: VDST = LDS byte address, VADDR = 64-bit
  // global address, no SADDR ("off"). Tracked with ASYNCcnt.
  asm volatile("global_load_async_to_lds_b32 %0, %1, off"
               :: "v"(dst), "v"(src) : "memory");
#else
  *(int*)lds_dst = *(const int*)gsrc;
#endif
}

__device__ __forceinline__ void wait_async_lds() {
#if defined(__gfx1250__)
  asm volatile("s_wait_asynccnt 0x0" ::: "memory");
#endif
}

// ---------------------------------------------------------------------------
// Kernel 1: one block per batch row. Hash-join ids2 -> ids, gather flow,
// accumulate sum(|xy - reordered|) and sum(mask) partials deterministically.
// ---------------------------------------------------------------------------
__global__ __launch_bounds__(NT) void track_loss_kernel(
    const float* __restrict__ flow,    // [B,2,H,W]
    const float* __restrict__ mask,    // [B,K]
    const int*   __restrict__ index,   // [B,K]
    const int*   __restrict__ ids,     // [B,K]
    const int*   __restrict__ index2,  // [B,K]
    const int*   __restrict__ ids2,    // [B,K]
    float*       __restrict__ part)    // [2*B] : [0..B) numer, [B..2B) mask
{
  __shared__ int   s_ids[K_];
  __shared__ int   s_idx[K_];
  __shared__ float s_msk[K_];
  __shared__ int   tbl_id[TBL];
  __shared__ int   tbl_pos[TBL];
  __shared__ float red_n[NT];
  __shared__ float red_m[NT];

  const int b   = blockIdx.x;
  const int tid = threadIdx.x;
  const long long row = (long long)b * K_;

  // clear hash table
  for (int i = tid; i < TBL; i += NT) tbl_id[i] = 0;

  // async-stage the three per-row arrays into LDS (ASYNCcnt-tracked)
  for (int i = tid; i < K_; i += NT)
    async_copy_dword(&s_ids[i], &ids[row + i]);
  for (int i = tid; i < K_; i += NT)
    async_copy_dword(&s_idx[i], &index[row + i]);
  for (int i = tid; i < K_; i += NT)
    async_copy_dword(&s_msk[i], &mask[row + i]);
  wait_async_lds();      // this wave's async loads landed in LDS
  __syncthreads();       // make them visible to all waves; tbl_id cleared

  // build hash table: id -> position x   (ids are unique & nonzero per row)
  float msum = 0.0f;
  for (int x = tid; x < K_; x += NT) {
    msum += s_msk[x];
    unsigned id = (unsigned)s_ids[x];
    unsigned h  = (id * 2654435761u) >> 22;   // Fibonacci hash -> 10 bits
    for (;;) {
      int old = atomicCAS(&tbl_id[h], 0, (int)id);
      if (old == 0) { tbl_pos[h] = x; break; }
      h = (h + 1) & (TBL - 1);
    }
  }
  __syncthreads();

  // probe: for each j find x with ids[x]==ids2[j]; accumulate |xy - vals|
  float acc = 0.0f;
  const float* fb = flow + (long long)b * (2 * HW_);
  for (int j = tid; j < K_; j += NT) {
    int id2 = ids2[row + j];
    if (id2 == 0) continue;                 // invalid
    unsigned h = ((unsigned)id2 * 2654435761u) >> 22;
    int x = -1;
    for (;;) {
      int v = tbl_id[h];
      if (v == 0) break;                    // unmatched (e.g. id2+100000)
      if (v == id2) { x = tbl_pos[h]; break; }
      h = (h + 1) & (TBL - 1);
    }
    if (x < 0) continue;

    int   gidx = s_idx[x];
    float m    = s_msk[x];
    float px   = fb[gidx]       * m;        // channel 0
    float py   = fb[HW_ + gidx] * m;        // channel 1
    int   i2   = index2[row + j];
    float vx   = (float)(i2 % W_);
    float vy   = (float)i2 / 272.0f;        // faithful: full index / 272.0
    acc += fabsf(px - vx) + fabsf(py - vy);
  }

  // deterministic block tree reduction (graph-replay safe, no float atomics)
  red_n[tid] = acc;
  red_m[tid] = msum;
  __syncthreads();
  for (int s = NT / 2; s > 0; s >>= 1) {
    if (tid < s) {
      red_n[tid] += red_n[tid + s];
      red_m[tid] += red_m[tid + s];
    }
    __syncthreads();
  }
  if (tid == 0) {
    part[b]      = red_n[0];
    part[B_ + b] = red_m[0];
  }
}

// ---------------------------------------------------------------------------
// Kernel 2: deterministic final reduction + normalization -> scalar output.
// ---------------------------------------------------------------------------
__global__ __launch_bounds__(NT) void finalize_kernel(
    const float* __restrict__ part, float* __restrict__ out)
{
  __shared__ float sn[NT];
  __shared__ float sm[NT];
  const int t = threadIdx.x;
  sn[t] = part[t];
  sm[t] = part[B_ + t];
  __syncthreads();
  for (int s = NT / 2; s > 0; s >>= 1) {
    if (t < s) { sn[t] += sn[t + s]; sm[t] += sm[t + s]; }
    __syncthreads();
  }
  if (t == 0) out[0] = sn[0] / (2.0f * sm[0] + 0.0001f);
}

// ---------------------------------------------------------------------------
extern "C" void kernel_launch(void* const* d_in, const int* in_sizes, int n_in,
                              void* d_out, int out_size, void* d_ws, size_t ws_size,
                              hipStream_t stream) {
  const float* flow   = (const float*)d_in[0];   // [B,2,H,W] f32
  const float* mask   = (const float*)d_in[1];   // [B,K]     f32
  const int*   index  = (const int*)  d_in[2];   // [B,K]     i32
  const int*   ids    = (const int*)  d_in[3];   // [B,K]     i32
  const int*   index2 = (const int*)  d_in[4];   // [B,K]     i32
  const int*   ids2   = (const int*)  d_in[5];   // [B,K]     i32

  float* part = (float*)d_ws;   // 2*B floats of scratch, fully overwritten

  track_loss_kernel<<<dim3(B_), dim3(NT), 0, stream>>>(
      flow, mask, index, ids, index2, ids2, part);
  finalize_kernel<<<dim3(1), dim3(NT), 0, stream>>>(part, (float*)d_out);
}